// cmodel_28200755265959
// MI455X (gfx1250) — compile-verified
//
#include <hip/hip_runtime.h>
#include <math.h>

typedef __attribute__((ext_vector_type(2))) float v2f;
typedef __attribute__((ext_vector_type(8))) float v8f;

#define EPSV 1e-5f

__device__ __forceinline__ float sigmf_(float x) { return 1.0f / (1.0f + __expf(-x)); }

__device__ __forceinline__ v8f wmma_f32_step(v2f a, v2f b2, v8f c) {
#if __has_builtin(__builtin_amdgcn_wmma_f32_16x16x4_f32)
    // D = A(16x4,f32) * B(4x16,f32) + C(16x16,f32)
    return __builtin_amdgcn_wmma_f32_16x16x4_f32(false, a, false, b2, (short)0, c,
                                                 false, false);
#else
    c[0] += a.x * b2.x + a.y * b2.y;  // unreachable fallback
    return c;
#endif
}

// Non-temporal loads for the giant streamed operands (main/side): keep the
// 192MB L2 free for the small reused tensors (xa, xm, xs).
__device__ __forceinline__ float ld_nt(const float* p) {
    return __builtin_nontemporal_load(p);
}
__device__ __forceinline__ v2f ld_nt2(const float* p) {
    return __builtin_nontemporal_load((const v2f*)p);
}

// ---------------------------------------------------------------------------
// Kernel 1: conv1d(3->3,k=3,pad=1) + BatchNorm(eval) + LeakyReLU(0.01)
// x: [1024,3,400] -> xa: [1024,3,400]
// ---------------------------------------------------------------------------
__global__ void conv_bn_lrelu(const float* __restrict__ x, const float* __restrict__ cw,
                              const float* __restrict__ cb, const float* __restrict__ bw,
                              const float* __restrict__ bb, const float* __restrict__ bm,
                              const float* __restrict__ bv, float* __restrict__ xa) {
    int idx = blockIdx.x * blockDim.x + threadIdx.x;
    const int total = 1024 * 3 * 400;
    if (idx >= total) return;
    int p = idx % 400;
    int o = (idx / 400) % 3;
    int b = idx / 1200;
    const float* xb = x + (size_t)b * 1200;
    float acc = cb[o];
#pragma unroll
    for (int i = 0; i < 3; ++i) {
#pragma unroll
        for (int dk = 0; dk < 3; ++dk) {
            int pp = p + dk - 1;
            if (pp >= 0 && pp < 400) acc += xb[i * 400 + pp] * cw[o * 9 + i * 3 + dk];
        }
    }
    acc = (acc - bm[o]) * rsqrtf(bv[o] + EPSV) * bw[o] + bb[o];
    xa[idx] = acc >= 0.f ? acc : 0.01f * acc;
}

// ---------------------------------------------------------------------------
// Kernel 2: batched C[b,3,N] = A[b,3,K] @ Bm[b,K,N], V_WMMA_F32_16X16X4_F32.
// One wave per (batch, 16-col tile). Branch-free inner loop:
//  - A rows >= 3 and B cols >= N are "don't care" (never stored) -> clamp
//    indices instead of masking; all loads unconditional.
//  - 32-bit element offsets from a per-batch base; A loaded as b64 pairs.
//  - Two independent accumulator chains (c0/c1) to pipeline WMMA latency.
//  - B stream loaded non-temporally (read-once, 0.9 GB total).
// ---------------------------------------------------------------------------
template <int K, int N>
__global__ void gemm_ab_wmma(const float* __restrict__ A, const float* __restrict__ Bm,
                             float* __restrict__ C) {
    const int b = blockIdx.y;
    const int n0 = blockIdx.x * 16;
    const int lane = threadIdx.x;
    const int half = lane >> 4;
    const int l16 = lane & 15;
    const int arow = (l16 < 3) ? l16 : 0;  // clamp: D rows 3..15 never stored
    const int n = n0 + l16;
    const int nc = (n < N) ? n : (N - 1);  // clamp: D cols >= N never stored
    const float* Arow = A + (size_t)b * 3 * K + (size_t)arow * K;
    const float* Bb = Bm + (size_t)b * K * N;
    v8f c0 = {}, c1 = {};
    int offB = half * 2 * N + nc;  // element offset of b.x at k0 = 0
#pragma unroll 2
    for (int k0 = 0; k0 < K; k0 += 8) {
        int ka = k0 + half * 2;
        v2f a0 = *(const v2f*)(Arow + ka);
        v2f a1 = *(const v2f*)(Arow + ka + 4);
        v2f b0, b1;
        b0.x = ld_nt(Bb + offB);
        b0.y = ld_nt(Bb + offB + N);
        b1.x = ld_nt(Bb + offB + 4 * N);
        b1.y = ld_nt(Bb + offB + 5 * N);
        offB += 8 * N;
        c0 = wmma_f32_step(a0, b0, c0);
        c1 = wmma_f32_step(a1, b1, c1);
    }
    if (lane < 16 && n < N) {
        v8f c = c0 + c1;
        float* Cb = C + (size_t)b * 3 * N;
        Cb[0 * N + n] = c[0];
        Cb[1 * N + n] = c[1];
        Cb[2 * N + n] = c[2];
    }
}

// ---------------------------------------------------------------------------
// Kernel 3 (fusion): out[b,3,400] = xm[b,3,150] @ main[b]^T + xs[b,3,400] @ side[b]^T
// Transposed-B access: B[k][n] = Mt[b,n,k], so each lane's k-stream is
// contiguous -> non-temporal b64 loads. K=150 handled with a 37-step main
// loop + one genuinely-masked tail step (K is reduced, so OOB k must be 0).
// ---------------------------------------------------------------------------
__global__ void gemm_fuse_wmma(const float* __restrict__ A1, const float* __restrict__ M1,
                               const float* __restrict__ A2, const float* __restrict__ M2,
                               float* __restrict__ C) {
    const int b = blockIdx.y;
    const int n0 = blockIdx.x * 16;  // N = 400, 25 exact tiles
    const int lane = threadIdx.x;
    const int half = lane >> 4;
    const int l16 = lane & 15;
    const int arow = (l16 < 3) ? l16 : 0;
    const int n = n0 + l16;
    v8f c0 = {}, c1 = {};
    {   // part 1: K = 150
        const float* A1r = A1 + (size_t)b * 450 + (size_t)arow * 150;
        const float* B1n = M1 + (size_t)b * 60000 + (size_t)n * 150;
        for (int k0 = 0; k0 < 148; k0 += 4) {  // k = 0..147
            int ka = k0 + half * 2;
            v2f av = *(const v2f*)(A1r + ka);
            v2f bv = ld_nt2(B1n + ka);
            c0 = wmma_f32_step(av, bv, c0);
        }
        // tail: k = 148,149 live in half 0; half 1 (k=150,151) must be zero
        v2f av = {0.f, 0.f}, bv = {0.f, 0.f};
        if (half == 0) {
            av = *(const v2f*)(A1r + 148);
            bv = ld_nt2(B1n + 148);
        }
        c0 = wmma_f32_step(av, bv, c0);
    }
    {   // part 2: K = 400
        const float* A2r = A2 + (size_t)b * 1200 + (size_t)arow * 400;
        const float* B2n = M2 + (size_t)b * 160000 + (size_t)n * 400;
#pragma unroll 2
        for (int k0 = 0; k0 < 400; k0 += 8) {
            int ka = k0 + half * 2;
            v2f a0 = *(const v2f*)(A2r + ka);
            v2f a1 = *(const v2f*)(A2r + ka + 4);
            v2f b0 = ld_nt2(B2n + ka);
            v2f b1 = ld_nt2(B2n + ka + 4);
            c0 = wmma_f32_step(a0, b0, c0);
            c1 = wmma_f32_step(a1, b1, c1);
        }
    }
    if (lane < 16) {
        v8f c = c0 + c1;
        float* Cb = C + (size_t)b * 1200;
        Cb[0 * 400 + n] = c[0];
        Cb[1 * 400 + n] = c[1];
        Cb[2 * 400 + n] = c[2];
    }
}

// ---------------------------------------------------------------------------
// Kernel 4: 2-layer LSTM (H=3, gate order i,f,g,o) + LayerNorm((T,H)).
// One wave per batch element; lanes 0..11 own one gate row each and activate
// it locally (tanh(x) = 2*sigmoid(2x)-1 unifies both activations branch-free,
// one exp per lane per step), then gates are exchanged via __shfl.
// ---------------------------------------------------------------------------
template <int T>
__global__ void lstm2_ln(const float* __restrict__ in, const float* __restrict__ Wih,
                         const float* __restrict__ Whh, const float* __restrict__ bih,
                         const float* __restrict__ bhh, const float* __restrict__ lnw,
                         const float* __restrict__ lnb, float* __restrict__ out) {
    __shared__ float sm[2][T * 3];
    const int b = blockIdx.x;
    const int lane = threadIdx.x;
    for (int idx = lane; idx < 3 * T; idx += 32) {
        int t = idx / 3, h = idx - t * 3;
        sm[0][idx] = in[(size_t)b * 3 * T + (size_t)h * T + t];
    }
    __syncthreads();
    int cur = 0;
    const int hid = lane < 3 ? lane : 0;
    const bool is_tanh_lane = (lane >= 6 && lane <= 8);  // 'g' gate rows
    const float kmul = is_tanh_lane ? 2.f : 1.f;
    const float sel_a = is_tanh_lane ? 2.f : 1.f;
    const float sel_b = is_tanh_lane ? -1.f : 0.f;
    for (int l = 0; l < 2; ++l) {
        const int j = lane < 12 ? lane : 0;
        const float wi0 = Wih[l * 36 + j * 3 + 0];
        const float wi1 = Wih[l * 36 + j * 3 + 1];
        const float wi2 = Wih[l * 36 + j * 3 + 2];
        const float wh0 = Whh[l * 36 + j * 3 + 0];
        const float wh1 = Whh[l * 36 + j * 3 + 1];
        const float wh2 = Whh[l * 36 + j * 3 + 2];
        const float bj = bih[l * 12 + j] + bhh[l * 12 + j];
        float h0 = 0.f, h1 = 0.f, h2 = 0.f, cc = 0.f;
        const float* bi = sm[cur];
        float* bo = sm[cur ^ 1];
        for (int t = 0; t < T; ++t) {
            float x0 = bi[t * 3 + 0], x1 = bi[t * 3 + 1], x2 = bi[t * 3 + 2];
            float g = bj + wi0 * x0 + wi1 * x1 + wi2 * x2 + wh0 * h0 + wh1 * h1 + wh2 * h2;
            float act = sel_a * sigmf_(kmul * g) + sel_b;  // sigmoid or tanh
            float gi = __shfl(act, hid + 0, 32);
            float gf = __shfl(act, hid + 3, 32);
            float gG = __shfl(act, hid + 6, 32);
            float go = __shfl(act, hid + 9, 32);
            cc = gf * cc + gi * gG;
            float hn = go * (2.f * sigmf_(2.f * cc) - 1.f);  // go * tanh(cc)
            h0 = __shfl(hn, 0, 32);
            h1 = __shfl(hn, 1, 32);
            h2 = __shfl(hn, 2, 32);
            if (lane < 3) bo[t * 3 + lane] = hn;
        }
        __syncthreads();
        cur ^= 1;
    }
    const float* v = sm[cur];
    float s = 0.f, s2 = 0.f;
    for (int idx = lane; idx < 3 * T; idx += 32) {
        float xv = v[idx];
        s += xv;
        s2 += xv * xv;
    }
#pragma unroll
    for (int off = 16; off > 0; off >>= 1) {
        s += __shfl_xor(s, off, 32);
        s2 += __shfl_xor(s2, off, 32);
    }
    const float inv_n = 1.0f / (float)(3 * T);
    float mu = s * inv_n;
    float var = s2 * inv_n - mu * mu;
    float rstd = rsqrtf(var + EPSV);
    for (int idx = lane; idx < 3 * T; idx += 32) {
        int t = idx / 3, h = idx - t * 3;
        out[(size_t)b * 3 * T + (size_t)h * T + t] = (v[idx] - mu) * rstd * lnw[idx] + lnb[idx];
    }
}

// ---------------------------------------------------------------------------
extern "C" void kernel_launch(void* const* d_in, const int* in_sizes, int n_in,
                              void* d_out, int out_size, void* d_ws, size_t ws_size,
                              hipStream_t stream) {
    (void)in_sizes; (void)n_in; (void)out_size; (void)ws_size;
    const float* x      = (const float*)d_in[0];
    const float* main_  = (const float*)d_in[1];
    const float* side   = (const float*)d_in[2];
    const float* conv_w = (const float*)d_in[3];
    const float* conv_b = (const float*)d_in[4];
    const float* bn_w   = (const float*)d_in[5];
    const float* bn_b   = (const float*)d_in[6];
    const float* bn_m   = (const float*)d_in[7];
    const float* bn_v   = (const float*)d_in[8];
    const float* mW[6], *sW[6], *aW[6];
    for (int i = 0; i < 6; ++i) {
        mW[i] = (const float*)d_in[9 + i];
        sW[i] = (const float*)d_in[15 + i];
        aW[i] = (const float*)d_in[21 + i];
    }

    // workspace layout (floats): xa | xm0 | xs0 | xall0  (~16.6 MB total)
    float* xa    = (float*)d_ws;
    float* xm0   = xa + (size_t)1024 * 1200;
    float* xs0   = xm0 + (size_t)1024 * 450;
    float* xall0 = xs0 + (size_t)1024 * 1200;

    // output layout: xm [B,3,150] | xs [B,3,400] | xall [B,3,400]
    float* xm_o   = (float*)d_out;
    float* xs_o   = xm_o + (size_t)1024 * 450;
    float* xall_o = xs_o + (size_t)1024 * 1200;

    conv_bn_lrelu<<<dim3((1024 * 1200 + 255) / 256), dim3(256), 0, stream>>>(
        x, conv_w, conv_b, bn_w, bn_b, bn_m, bn_v, xa);

    gemm_ab_wmma<400, 150><<<dim3(10, 1024), dim3(32), 0, stream>>>(xa, main_, xm0);
    gemm_ab_wmma<400, 400><<<dim3(25, 1024), dim3(32), 0, stream>>>(xa, side, xs0);

    lstm2_ln<150><<<dim3(1024), dim3(32), 0, stream>>>(xm0, mW[0], mW[1], mW[2], mW[3],
                                                       mW[4], mW[5], xm_o);
    lstm2_ln<400><<<dim3(1024), dim3(32), 0, stream>>>(xs0, sW[0], sW[1], sW[2], sW[3],
                                                       sW[4], sW[5], xs_o);

    gemm_fuse_wmma<<<dim3(25, 1024), dim3(32), 0, stream>>>(xm_o, main_, xs_o, side, xall0);

    lstm2_ln<400><<<dim3(1024), dim3(32), 0, stream>>>(xall0, aW[0], aW[1], aW[2], aW[3],
                                                       aW[4], aW[5], xall_o);
}